// YoloDetectionLoss_59846074302840
// MI455X (gfx1250) — compile-verified
//
#include <hip/hip_runtime.h>
#include <hip/hip_bf16.h>
#include <math.h>

typedef __attribute__((ext_vector_type(2))) float v2f;
typedef __attribute__((ext_vector_type(8))) float v8f;

#define NUM_CLASSES 80
#define A_PER 3
#define IMG 416.0f
#define BOX_W 1.0f
#define OBJ_W 1.0f
#define NOOBJ_W 0.25f
#define CLS_W 0.5f
#define BATCH 32
#define NGT 60

// cells per scale: 32*3*13^2, 32*3*26^2, 32*3*52^2
#define CS0 16224
#define CS1 64896
#define CS2 259584
#define NCELLS (CS0 + CS1 + CS2)

__constant__ float c_anchors[9][2] = {
    {116.f, 90.f}, {156.f, 198.f}, {373.f, 326.f},
    {30.f, 61.f},  {62.f, 45.f},   {59.f, 119.f},
    {10.f, 13.f},  {16.f, 30.f},   {33.f, 23.f}};

// ---- CDNA5 wave32 reduction using V_WMMA_F32_16X16X4_F32 with B = ones ----
// A layout (16x4 f32): lanes 0-15 hold K=0 (vgpr0) / K=1 (vgpr1),
// lanes 16-31 hold K=2 / K=3.  With a={v,0}: rowsum(m) = v(m) + v(m+16).
// D layout: vgpr r, lanes 0-15 -> M=r, lanes 16-31 -> M=r+8, value equal for
// all N.  Sum of d[0..7] gives half-wave sum; shfl_xor(16) completes it.
// EXEC must be all ones at the call site (callers keep full blocks converged).
__device__ __forceinline__ float wave_reduce_wmma(float v) {
  v2f a;
  a[0] = v;
  a[1] = 0.0f;
  v2f b;
  b[0] = 1.0f;
  b[1] = 1.0f;
  v8f c = {};
  c = __builtin_amdgcn_wmma_f32_16x16x4_f32(false, a, false, b, (short)0, c,
                                            false, false);
  float s = c[0] + c[1] + c[2] + c[3] + c[4] + c[5] + c[6] + c[7];
  s += __shfl_xor(s, 16, 32);
  return s;
}

__device__ __forceinline__ float bce_neg(float x) {  // BCE(x, y=0)
  return fmaxf(x, 0.0f) + log1pf(expf(-fabsf(x)));
}

// ---- Pass 0: zero accumulators + dedup flags in workspace ----
__global__ void yolo_zero(unsigned* __restrict__ ws, int nwords) {
  int i = blockIdx.x * blockDim.x + threadIdx.x;
  if (i < nwords) ws[i] = 0u;
}

// ---- Pass 1: anchor assignment + positive-cell contributions ----
// acc layout per scale s (8 floats): [0]=n_pos [1]=S_box [2]=S_obj_pos
// [3]=S_obj_neg (corrections here, sweep adds later) [4]=S_cls
__global__ void yolo_assign(const float* __restrict__ p0,
                            const float* __restrict__ p1,
                            const float* __restrict__ p2,
                            const float* __restrict__ tgt,
                            float* __restrict__ acc,
                            unsigned* __restrict__ flags) {
  int i = blockIdx.x * blockDim.x + threadIdx.x;
  if (i >= BATCH * NGT) return;
  int b = i / NGT;
  const float* t = tgt + (size_t)i * 5;
  float cidf = t[0], cx = t[1], cy = t[2], w = t[3], h = t[4];

  // best anchor by wh-IoU over the 9 flat anchors (first max wins, like argmax)
  float gw = w * IMG, gh = h * IMG;
  int best = 0;
  float bestIou = -1.0f;
#pragma unroll
  for (int j = 0; j < 9; j++) {
    float aw = c_anchors[j][0], ah = c_anchors[j][1];
    float inter = fminf(gw, aw) * fminf(gh, ah);
    float uni = gw * gh + aw * ah - inter;
    float iou = inter / fmaxf(uni, 1e-6f);
    if (iou > bestIou) {
      bestIou = iou;
      best = j;
    }
  }
  int s = best / 3, a = best - 3 * s;
  int cid = min(max((int)cidf, 0), NUM_CLASSES - 1);

  int W = 13 << s, H = W, HW = W * W;
  float gx = cx * (float)W, gy = cy * (float)H;
  int gxi = min(max((int)floorf(gx), 0), W - 1);
  int gyi = min(max((int)floorf(gy), 0), H - 1);

  int cellLocal = ((b * A_PER + a) * H + gyi) * W + gxi;
  int cellBase = (s == 0) ? 0 : ((s == 1) ? CS0 : (CS0 + CS1));

  // scatter-set dedup: one representative GT per cell
  unsigned old = atomicExch(&flags[cellBase + cellLocal], 1u);
  if (old) return;

  const float* pred = (s == 0) ? p0 : ((s == 1) ? p1 : p2);
  const float* cell =
      pred + (size_t)(b * (A_PER * (5 + NUM_CLASSES)) + a * (5 + NUM_CLASSES)) * HW +
      (size_t)gyi * W + gxi;  // channel c at cell[c*HW]

  float px = cell[0 * HW], py = cell[1 * (size_t)HW];
  float pw = cell[2 * (size_t)HW], ph = cell[3 * (size_t)HW];
  float po = cell[4 * (size_t)HW];

  float tx = gx - (float)gxi, ty = gy - (float)gyi;
  float sfac = (float)W / IMG;
  float anw = c_anchors[s * 3 + a][0] * sfac;
  float anh = c_anchors[s * 3 + a][1] * sfac;
  float twl = fminf(fmaxf(logf(fmaxf(w * W, 1e-6f) / fmaxf(anw, 1e-6f)), -6.f), 6.f);
  float thl = fminf(fmaxf(logf(fmaxf(h * H, 1e-6f) / fmaxf(anh, 1e-6f)), -6.f), 6.f);

  float sx = 1.0f / (1.0f + expf(-px));
  float sy = 1.0f / (1.0f + expf(-py));
  float lbox = (sx - tx) * (sx - tx) + (sy - ty) * (sy - ty) +
               (pw - twl) * (pw - twl) + (ph - thl) * (ph - thl);

  // class cross-entropy via stable log-softmax over 80 strided channels
  const float* cl = cell + 5 * (size_t)HW;
  float m = -3.4e38f;
  for (int c = 0; c < NUM_CLASSES; c++) m = fmaxf(m, cl[(size_t)c * HW]);
  float se = 0.0f;
  for (int c = 0; c < NUM_CLASSES; c++) se += expf(cl[(size_t)c * HW] - m);
  float ce = (m + logf(se)) - cl[(size_t)cid * HW];

  float b0 = bce_neg(po);        // BCE(po, 0) — to be removed from neg sum
  float b1 = b0 - po;            // BCE(po, 1)

  float* ac = acc + s * 8;
  atomicAdd(&ac[0], 1.0f);
  atomicAdd(&ac[1], lbox);
  atomicAdd(&ac[2], b1);
  atomicAdd(&ac[3], -b0);        // sweep counts this cell as negative; correct it
  atomicAdd(&ac[4], ce);
}

// ---- Pass 2: objectness BCE sweep over every cell (as-if-negative) ----
__global__ void yolo_obj_sweep(const float* __restrict__ p0,
                               const float* __restrict__ p1,
                               const float* __restrict__ p2,
                               float* __restrict__ acc) {
  float accS[3] = {0.f, 0.f, 0.f};
  int stride = gridDim.x * blockDim.x;
  for (int idx = blockIdx.x * blockDim.x + threadIdx.x; idx < NCELLS;
       idx += stride) {
    int s, local;
    if (idx < CS0) {
      s = 0;
      local = idx;
    } else if (idx < CS0 + CS1) {
      s = 1;
      local = idx - CS0;
    } else {
      s = 2;
      local = idx - (CS0 + CS1);
    }
    int W = 13 << s, HW = W * W;
    int b = local / (A_PER * HW);
    int r = local - b * A_PER * HW;
    int a = r / HW;
    int pix = r - a * HW;
    const float* pred = (s == 0) ? p0 : ((s == 1) ? p1 : p2);
    float x = pred[(size_t)(b * (A_PER * (5 + NUM_CLASSES)) +
                            a * (5 + NUM_CLASSES) + 4) * HW + pix];
    accS[s] += bce_neg(x);
  }

  // wave reduce via WMMA (converged: full block, uniform loop exit),
  // then LDS across the 8 waves of the block, then 3 atomics per block.
  __shared__ float red[8][3];
  int wave = threadIdx.x >> 5;
#pragma unroll
  for (int s = 0; s < 3; s++) {
    float ws = wave_reduce_wmma(accS[s]);
    if ((threadIdx.x & 31) == 0) red[wave][s] = ws;
  }
  __syncthreads();
  if (threadIdx.x == 0) {
#pragma unroll
    for (int s = 0; s < 3; s++) {
      float tsum = 0.0f;
      for (int w = 0; w < 8; w++) tsum += red[w][s];
      atomicAdd(&acc[s * 8 + 3], tsum);
    }
  }
}

// ---- Pass 3: combine per-scale sums into the scalar loss ----
__global__ void yolo_finalize(const float* __restrict__ acc,
                              float* __restrict__ out) {
  if (blockIdx.x == 0 && threadIdx.x == 0) {
    const float tot[3] = {(float)CS0, (float)CS1, (float)CS2};
    float loss = 0.0f;
    for (int s = 0; s < 3; s++) {
      const float* ac = acc + s * 8;
      float np = fmaxf(ac[0], 1.0f);
      float nn = fmaxf(tot[s] - ac[0], 1.0f);
      loss += BOX_W * ac[1] / np + OBJ_W * ac[2] / np + NOOBJ_W * ac[3] / nn +
              CLS_W * ac[4] / np;
    }
    out[0] = loss;
  }
}

extern "C" void kernel_launch(void* const* d_in, const int* in_sizes, int n_in,
                              void* d_out, int out_size, void* d_ws,
                              size_t ws_size, hipStream_t stream) {
  const float* p0 = (const float*)d_in[0];
  const float* p1 = (const float*)d_in[1];
  const float* p2 = (const float*)d_in[2];
  const float* tg = (const float*)d_in[3];

  // workspace: 32 accumulator floats, then NCELLS u32 dedup flags
  float* acc = (float*)d_ws;
  unsigned* flags = (unsigned*)d_ws + 32;
  const int nwords = 32 + NCELLS;

  yolo_zero<<<(nwords + 255) / 256, 256, 0, stream>>>((unsigned*)d_ws, nwords);
  yolo_assign<<<(BATCH * NGT + 255) / 256, 256, 0, stream>>>(p0, p1, p2, tg,
                                                             acc, flags);
  yolo_obj_sweep<<<512, 256, 0, stream>>>(p0, p1, p2, acc);
  yolo_finalize<<<1, 64, 0, stream>>>(acc, (float*)d_out);
}